// ChannelAttention_30073361007272
// MI455X (gfx1250) — compile-verified
//
#include <hip/hip_runtime.h>

typedef __attribute__((ext_vector_type(2))) float v2f;
typedef __attribute__((ext_vector_type(8))) float v8f;

#define BSZ 16
#define CDIM 256
#define SDIM 1024

// Kernel 1: pooled[b*C+c] = mean_s x[b,c,s].  One block per (b,c) row,
// 256 threads * one float4 each = 1024 floats, LDS tree reduction.
__global__ __launch_bounds__(256) void pool_kernel(const float* __restrict__ x,
                                                   float* __restrict__ pooled) {
    __shared__ float red[256];
    const int row = blockIdx.x;                       // b*C + c
    const float4* p = (const float4*)(x + (size_t)row * SDIM);
    float4 v = p[threadIdx.x];
    red[threadIdx.x] = v.x + v.y + v.z + v.w;
    __syncthreads();
    for (int off = 128; off > 0; off >>= 1) {
        if (threadIdx.x < off) red[threadIdx.x] += red[threadIdx.x + off];
        __syncthreads();
    }
    if (threadIdx.x == 0) pooled[row] = red[0] * (1.0f / SDIM);
}

// Kernel 2: gate = (pooled @ wv.T + bv) @ out_proj_w.T + out_proj_b
// Single workgroup, 16 waves. Wave w owns output columns [16w, 16w+16).
// M=16 (batch) fits exactly one WMMA tile; K=256 swept in 64 steps of K=4
// with V_WMMA_F32_16X16X4_F32 (full fp32).
//
// A layout (16x4 f32): lane m=lane&15 holds row m; a.x=K(k0+2*half),
// a.y=K(k0+2*half+1) with half=lane>>4.
// B layout (4x16 f32): B[k][n] = W[n][k] -> contiguous-in-k 8B load.
// C/D layout: VGPR r -> M = r + 8*half, N = lane&15 (within wave's slice).
__global__ __launch_bounds__(512) void gate_kernel(const float* __restrict__ pooled,
                                                   const float* __restrict__ in_proj_w,
                                                   const float* __restrict__ in_proj_b,
                                                   const float* __restrict__ out_proj_w,
                                                   const float* __restrict__ out_proj_b,
                                                   float* __restrict__ gate) {
    __shared__ float tmpS[BSZ * CDIM];                // 16 KB intermediate [16][256]
    const int lane = threadIdx.x & 31;
    const int wave = threadIdx.x >> 5;
    const int m    = lane & 15;
    const int half = lane >> 4;
    const int n    = wave * 16 + m;                   // output column this lane handles (B/C/D)

    const float* wv  = in_proj_w + 2 * CDIM * CDIM;   // rows 512..767 of in_proj_w
    const float* bv  = in_proj_b + 2 * CDIM;

    // Stage A: tmp = pooled @ wv.T  (+ bv)
    v8f acc = {};
    for (int k0 = 0; k0 < CDIM; k0 += 4) {
        const int ka = k0 + half * 2;
        v2f a = *(const v2f*)(pooled + m * CDIM + ka);
        v2f b = *(const v2f*)(wv + n * CDIM + ka);
        acc = __builtin_amdgcn_wmma_f32_16x16x4_f32(false, a, false, b,
                                                    (short)0, acc, false, false);
    }
    {
        const float bn = bv[n];
#pragma unroll
        for (int r = 0; r < 8; ++r)
            tmpS[(r + half * 8) * CDIM + n] = acc[r] + bn;
    }
    __syncthreads();

    // Stage B: gate = tmp @ out_proj_w.T (+ out_proj_b)
    v8f acc2 = {};
    for (int k0 = 0; k0 < CDIM; k0 += 4) {
        const int ka = k0 + half * 2;
        v2f a = *(const v2f*)(tmpS + m * CDIM + ka);  // ds_load_b64
        v2f b = *(const v2f*)(out_proj_w + n * CDIM + ka);
        acc2 = __builtin_amdgcn_wmma_f32_16x16x4_f32(false, a, false, b,
                                                     (short)0, acc2, false, false);
    }
    {
        const float bo = out_proj_b[n];
#pragma unroll
        for (int r = 0; r < 8; ++r)
            gate[(r + half * 8) * CDIM + n] = acc2[r] + bo;
    }
}

// Kernel 3: out[b,c,s] = x[b,c,s] * gate[b*C+c].  float4 vectorized;
// 256 float4 per row so all 4 lanes of a vector share one gate value.
__global__ __launch_bounds__(256) void mul_kernel(const float* __restrict__ x,
                                                  const float* __restrict__ gate,
                                                  float* __restrict__ out) {
    const size_t idx = (size_t)blockIdx.x * 256 + threadIdx.x;   // float4 index
    const int row = (int)(idx >> 8);                             // /(1024/4)
    const float g = gate[row];
    float4 v = ((const float4*)x)[idx];
    v.x *= g; v.y *= g; v.z *= g; v.w *= g;
    ((float4*)out)[idx] = v;
}

extern "C" void kernel_launch(void* const* d_in, const int* in_sizes, int n_in,
                              void* d_out, int out_size, void* d_ws, size_t ws_size,
                              hipStream_t stream) {
    const float* x          = (const float*)d_in[0];   // [16,256,1024]
    const float* in_proj_w  = (const float*)d_in[1];   // [768,256]
    const float* in_proj_b  = (const float*)d_in[2];   // [768]
    const float* out_proj_w = (const float*)d_in[3];   // [256,256]
    const float* out_proj_b = (const float*)d_in[4];   // [256]
    float* out = (float*)d_out;                        // [16,256,1024]

    float* pooled = (float*)d_ws;                      // 4096 floats
    float* gate   = pooled + BSZ * CDIM;               // 4096 floats

    pool_kernel<<<BSZ * CDIM, 256, 0, stream>>>(x, pooled);
    gate_kernel<<<1, 512, 0, stream>>>(pooled, in_proj_w, in_proj_b,
                                       out_proj_w, out_proj_b, gate);
    mul_kernel<<<(BSZ * CDIM * SDIM) / 4 / 256, 256, 0, stream>>>(x, gate, out);

    (void)in_sizes; (void)n_in; (void)out_size; (void)ws_size;
}